// SelfAttention_45200235823180
// MI455X (gfx1250) — compile-verified
//
#include <hip/hip_runtime.h>

#define B_ 4
#define S_ 2048
#define D_ 256
#define H_ 8
#define DH_ 32

typedef __attribute__((ext_vector_type(2))) float v2f;
typedef __attribute__((ext_vector_type(8))) float v8f;

// ---- workspace layout (float offsets into d_ws) ----
enum : int {
  OFF_POOLQ = 0,                       // B*D  (sum of Q rows)
  OFF_POOLV = OFF_POOLQ + B_ * D_,     // B*D  (sum of V rows)
  OFF_SQ    = OFF_POOLV + B_ * D_,     // B*D  (mq-masked sum of Q rows)
  OFF_PV    = OFF_SQ    + B_ * D_,     // B*H*D (softmax-weighted V sums)
  OFF_TT    = OFF_PV    + B_ * H_ * D_,// B*D*16 (T transposed, head-padded)
  ZERO_CNT  = OFF_TT    + B_ * D_ * 16,
  OFF_GK    = ZERO_CNT,                // B*D gate-k vector
  OFF_GQ    = OFF_GK + B_ * D_,        // B*D gate-q vector
  OFF_CKQ   = OFF_GQ + B_ * D_,        // 2*B gate scalars
  OFF_OACT  = OFF_CKQ + 2 * B_,        // B*D
  OFF_OUNIF = OFF_OACT + B_ * D_,      // B*D
  OFF_MASK  = OFF_OUNIF + B_ * D_,     // uchar region: 2*B*S bytes
};

__device__ __forceinline__ v8f wmma_f32_4(v2f a, v2f b, v8f c) {
  // V_WMMA_F32_16X16X4_F32 : D = A(16x4,f32) * B(4x16,f32) + C(16x16,f32)
  return __builtin_amdgcn_wmma_f32_16x16x4_f32(false, a, false, b, (short)0, c,
                                               false, false);
}

__global__ void k_zero(float* w) {
  int i = blockIdx.x * 256 + threadIdx.x;
  if (i < ZERO_CNT) w[i] = 0.f;
}

// ---- pass 1: column sums of Q and V (for pool / uniform output) ----
__global__ void k_pool(const float* __restrict__ Q, const float* __restrict__ V,
                       float* w) {
  int b = blockIdx.y, t = threadIdx.x;
  int s0 = blockIdx.x * 128;
  const float* qp = Q + ((size_t)b * S_ + s0) * D_ + t;
  const float* vp = V + ((size_t)b * S_ + s0) * D_ + t;
  float aq = 0.f, av = 0.f;
  for (int i = 0; i < 128; ++i) { aq += qp[(size_t)i * D_]; av += vp[(size_t)i * D_]; }
  atomicAdd(&w[OFF_POOLQ + b * D_ + t], aq);
  atomicAdd(&w[OFF_POOLV + b * D_ + t], av);
}

// ---- collapse the gate MLP: gk = wa + pool*wc ; ck = pool.wb + b1.W2 + b2 ----
__global__ void k_prepg(const float* __restrict__ W1k, const float* __restrict__ b1k,
                        const float* __restrict__ W2k, const float* __restrict__ b2k,
                        const float* __restrict__ W1q, const float* __restrict__ b1q,
                        const float* __restrict__ W2q, const float* __restrict__ b2q,
                        float* w) {
  int b = blockIdx.x, t = threadIdx.x;
  __shared__ float w2k_s[D_], w2q_s[D_], pool_s[D_], red[256];
  w2k_s[t] = W2k[t];
  w2q_s[t] = W2q[t];
  pool_s[t] = w[OFF_POOLQ + b * D_ + t] * (1.f / S_);
  __syncthreads();
  const float* r0k = W1k + (size_t)t * D_;
  const float* r1k = W1k + (size_t)(256 + t) * D_;
  const float* r2k = W1k + (size_t)(512 + t) * D_;
  const float* r0q = W1q + (size_t)t * D_;
  const float* r1q = W1q + (size_t)(256 + t) * D_;
  const float* r2q = W1q + (size_t)(512 + t) * D_;
  float ak = 0, bk = 0, ck = 0, aq = 0, bq = 0, cq = 0;
  for (int d = 0; d < D_; ++d) {
    float xk = w2k_s[d], xq = w2q_s[d];
    ak += r0k[d] * xk; bk += r1k[d] * xk; ck += r2k[d] * xk;
    aq += r0q[d] * xq; bq += r1q[d] * xq; cq += r2q[d] * xq;
  }
  w[OFF_GK + b * D_ + t] = ak + pool_s[t] * ck;
  w[OFF_GQ + b * D_ + t] = aq + pool_s[t] * cq;
  red[t] = pool_s[t] * bk + b1k[t] * w2k_s[t];
  __syncthreads();
  for (int o = 128; o; o >>= 1) { if (t < o) red[t] += red[t + o]; __syncthreads(); }
  if (t == 0) w[OFF_CKQ + b * 2 + 0] = red[0] + b2k[0];
  __syncthreads();
  red[t] = pool_s[t] * bq + b1q[t] * w2q_s[t];
  __syncthreads();
  for (int o = 128; o; o >>= 1) { if (t < o) red[t] += red[t + o]; __syncthreads(); }
  if (t == 0) w[OFF_CKQ + b * 2 + 1] = red[0] + b2q[0];
}

// ---- pass 2: per-token gates + mq-masked Q row sum (LDS ds_add_f32 atomics) ----
__global__ void k_gate(const float* __restrict__ Q, float* w) {
  int b = blockIdx.y, chunk = blockIdx.x, t = threadIdx.x;
  int wv = t >> 5, l = t & 31;
  __shared__ float gk_s[D_], gq_s[D_], sql[D_];
  gk_s[t] = w[OFF_GK + b * D_ + t];
  gq_s[t] = w[OFF_GQ + b * D_ + t];
  sql[t] = 0.f;
  __syncthreads();
  float ckb = w[OFF_CKQ + b * 2 + 0];
  float cqb = w[OFF_CKQ + b * 2 + 1];
  unsigned char* mk = (unsigned char*)(w + OFF_MASK);
  unsigned char* mq = mk + B_ * S_;
  float a0 = 0, a1 = 0, a2 = 0, a3 = 0, a4 = 0, a5 = 0, a6 = 0, a7 = 0;
  for (int i = 0; i < 32; ++i) {
    int s = chunk * 256 + wv * 32 + i;
    const float4* qr = (const float4*)(Q + ((size_t)b * S_ + s) * D_);
    float4 x0 = qr[l * 2], x1 = qr[l * 2 + 1];
    float dk = x0.x * gk_s[l * 8 + 0] + x0.y * gk_s[l * 8 + 1] +
               x0.z * gk_s[l * 8 + 2] + x0.w * gk_s[l * 8 + 3] +
               x1.x * gk_s[l * 8 + 4] + x1.y * gk_s[l * 8 + 5] +
               x1.z * gk_s[l * 8 + 6] + x1.w * gk_s[l * 8 + 7];
    float dq = x0.x * gq_s[l * 8 + 0] + x0.y * gq_s[l * 8 + 1] +
               x0.z * gq_s[l * 8 + 2] + x0.w * gq_s[l * 8 + 3] +
               x1.x * gq_s[l * 8 + 4] + x1.y * gq_s[l * 8 + 5] +
               x1.z * gq_s[l * 8 + 6] + x1.w * gq_s[l * 8 + 7];
    for (int m = 16; m; m >>= 1) { dk += __shfl_xor(dk, m); dq += __shfl_xor(dq, m); }
    int mkb = (dk + ckb) > 0.f;   // sigmoid(x)>0.5  <=>  x>0
    int mqb = (dq + cqb) > 0.f;
    if (l == 0) { mk[b * S_ + s] = (unsigned char)mkb; mq[b * S_ + s] = (unsigned char)mqb; }
    if (mqb) { a0 += x0.x; a1 += x0.y; a2 += x0.z; a3 += x0.w;
               a4 += x1.x; a5 += x1.y; a6 += x1.z; a7 += x1.w; }
  }
  atomicAdd(&sql[l * 8 + 0], a0); atomicAdd(&sql[l * 8 + 1], a1);
  atomicAdd(&sql[l * 8 + 2], a2); atomicAdd(&sql[l * 8 + 3], a3);
  atomicAdd(&sql[l * 8 + 4], a4); atomicAdd(&sql[l * 8 + 5], a5);
  atomicAdd(&sql[l * 8 + 6], a6); atomicAdd(&sql[l * 8 + 7], a7);
  __syncthreads();
  atomicAdd(&w[OFF_SQ + b * D_ + t], sql[t]);
}

// ---- tiny: qbar = sq@WQ_h ; Tt[b,d,h] = scale * WK_h @ qbar ----
__global__ void k_qbarT(const float* __restrict__ WQ, const float* __restrict__ WK,
                        float* w) {
  int b = blockIdx.x >> 3, h = blockIdx.x & 7, t = threadIdx.x;
  __shared__ float sq_s[D_], part[256], qb[32];
  sq_s[t] = w[OFF_SQ + b * D_ + t];
  __syncthreads();
  int j = t & 31, sl = t >> 5;
  float p = 0.f;
  for (int dd = 0; dd < 32; ++dd) {
    int d = sl * 32 + dd;
    p += sq_s[d] * WQ[(size_t)d * D_ + h * DH_ + j];
  }
  part[t] = p;
  __syncthreads();
  if (t < 32) {
    float s = 0.f;
    for (int k2 = 0; k2 < 8; ++k2) s += part[t + 32 * k2];
    qb[t] = s;
  }
  __syncthreads();
  const float scale = 0.17677669529663687f;  // 1/sqrt(32)
  float tv = 0.f;
  for (int j2 = 0; j2 < 32; ++j2) tv += WK[(size_t)t * D_ + h * DH_ + j2] * qb[j2];
  w[OFF_TT + (b * D_ + t) * 16 + h] = tv * scale;
}

// ---- pass 3 (WMMA): R[b,h,s] = K_seq[b] @ Tt[b]  (M=S tiles of 16, N=16, K=256) ----
__global__ void k_logits(const float* __restrict__ K, const float* __restrict__ w,
                         float* __restrict__ Rbuf) {
  int b = blockIdx.y;
  int wv = threadIdx.x >> 5, l = threadIdx.x & 31;
  int m0 = blockIdx.x * 128 + wv * 16;
  int half = l >> 4, n = l & 15;
  // A 16x4 f32 layout: lane holds M=lane%16, K = k0 + 2*(lane/16) + {0,1}
  const v2f* arow = (const v2f*)(K + ((size_t)b * S_ + m0 + n) * D_);
  const float* tt = w + OFF_TT + b * D_ * 16;
  v8f c = {0.f, 0.f, 0.f, 0.f, 0.f, 0.f, 0.f, 0.f};
  for (int k0 = 0; k0 < D_; k0 += 4) {
    v2f a = arow[(k0 >> 1) + half];
    int kk = k0 + 2 * half;
    v2f bv;
    bv.x = tt[kk * 16 + n];
    bv.y = tt[(kk + 1) * 16 + n];
    c = wmma_f32_4(a, bv, c);
  }
  if (n < H_) {  // C/D layout: VGPR v holds M=v+8*half, N=lane%16
    float* rr = Rbuf + ((size_t)b * 16 + n) * S_ + m0 + 8 * half;
#pragma unroll
    for (int v = 0; v < 8; ++v) rr[v] = c[v];
  }
}

// ---- softmax over S per (b,h); pad heads 8..15 with zeros ----
__global__ void k_softmax(const float* __restrict__ Rbuf, float* __restrict__ Pbuf) {
  int b = blockIdx.x >> 4, h = blockIdx.x & 15, t = threadIdx.x;
  float* Pr = Pbuf + ((size_t)b * 16 + h) * S_;
  if (h >= H_) {
    for (int s = t; s < S_; s += 256) Pr[s] = 0.f;
    return;
  }
  const float* Rr = Rbuf + ((size_t)b * 16 + h) * S_;
  __shared__ float red[256];
  float mx = -3.4e38f;
  for (int s = t; s < S_; s += 256) mx = fmaxf(mx, Rr[s]);
  red[t] = mx;
  __syncthreads();
  for (int o = 128; o; o >>= 1) { if (t < o) red[t] = fmaxf(red[t], red[t + o]); __syncthreads(); }
  mx = red[0];
  __syncthreads();
  float sm = 0.f;
  for (int s = t; s < S_; s += 256) sm += __expf(Rr[s] - mx);
  red[t] = sm;
  __syncthreads();
  for (int o = 128; o; o >>= 1) { if (t < o) red[t] += red[t + o]; __syncthreads(); }
  float inv = 1.f / red[0];
  for (int s = t; s < S_; s += 256) Pr[s] = __expf(Rr[s] - mx) * inv;
}

// ---- pass 4 (WMMA): PV[b,h,d] = P[b,16,S] @ V_seq[b,S,256], K split over S ----
__global__ void k_pv(const float* __restrict__ V, const float* __restrict__ Pbuf,
                     float* w) {
  int g = blockIdx.x * 8 + (threadIdx.x >> 5);
  int l = threadIdx.x & 31;
  int b = g >> 7, nt = (g >> 3) & 15, sc = g & 7;
  int d0 = nt * 16, s0 = sc * 256;
  int half = l >> 4, n = l & 15;
  const v2f* prow = (const v2f*)(Pbuf + ((size_t)b * 16 + n) * S_);  // A: M=lane%16
  v8f c = {0.f, 0.f, 0.f, 0.f, 0.f, 0.f, 0.f, 0.f};
  for (int k0 = s0; k0 < s0 + 256; k0 += 4) {
    v2f a = prow[(k0 >> 1) + half];
    int kk = k0 + 2 * half;
    v2f bv;
    bv.x = V[((size_t)b * S_ + kk) * D_ + d0 + n];
    bv.y = V[((size_t)b * S_ + kk + 1) * D_ + d0 + n];
    c = wmma_f32_4(a, bv, c);
  }
  if (half == 0) {  // rows M=0..7 are the real heads
#pragma unroll
    for (int v = 0; v < 8; ++v)
      atomicAdd(&w[OFF_PV + ((size_t)b * H_ + v) * D_ + d0 + n], c[v]);
  }
}

// ---- tiny: O_active = pv@WV_h ; O_uniform = mean(V)@WV ----
__global__ void k_oprep(const float* __restrict__ WV, float* w) {
  int b = blockIdx.x, e = threadIdx.x;
  int h = e >> 5;
  float oa = 0.f, ou = 0.f;
  for (int d = 0; d < D_; ++d) {
    float x = WV[(size_t)d * D_ + e];
    oa += w[OFF_PV + (b * H_ + h) * D_ + d] * x;
    ou += w[OFF_POOLV + b * D_ + d] * x;
  }
  w[OFF_OACT + b * D_ + e] = oa;
  w[OFF_OUNIF + b * D_ + e] = ou * (1.f / S_);
}

// ---- pass 5: broadcast-select rows into the output ----
__global__ void k_write(const float* __restrict__ w, float* __restrict__ out) {
  int b = blockIdx.y, t = threadIdx.x;
  int s0 = blockIdx.x * 8;
  float oa = w[OFF_OACT + b * D_ + t];
  float ou = w[OFF_OUNIF + b * D_ + t];
  const unsigned char* mk = (const unsigned char*)(w + OFF_MASK);
  for (int i = 0; i < 8; ++i) {
    int s = s0 + i;
    out[((size_t)b * S_ + s) * D_ + t] = mk[b * S_ + s] ? oa : ou;
  }
}

extern "C" void kernel_launch(void* const* d_in, const int* in_sizes, int n_in,
                              void* d_out, int out_size, void* d_ws, size_t ws_size,
                              hipStream_t stream) {
  const float* Q   = (const float*)d_in[0];
  const float* K   = (const float*)d_in[1];
  const float* V   = (const float*)d_in[2];
  const float* WQ  = (const float*)d_in[3];
  const float* WK  = (const float*)d_in[4];
  const float* WV  = (const float*)d_in[5];
  const float* W1k = (const float*)d_in[6];
  const float* b1k = (const float*)d_in[7];
  const float* W2k = (const float*)d_in[8];
  const float* b2k = (const float*)d_in[9];
  const float* W1q = (const float*)d_in[10];
  const float* b1q = (const float*)d_in[11];
  const float* W2q = (const float*)d_in[12];
  const float* b2q = (const float*)d_in[13];
  float* out = (float*)d_out;
  float* w = (float*)d_ws;
  // R and P scratch live inside d_out; k_write overwrites d_out last.
  float* Rbuf = out;
  float* Pbuf = out + B_ * 16 * S_;

  k_zero<<<dim3((ZERO_CNT + 255) / 256), dim3(256), 0, stream>>>(w);
  k_pool<<<dim3(16, B_), dim3(256), 0, stream>>>(Q, V, w);
  k_prepg<<<dim3(B_), dim3(256), 0, stream>>>(W1k, b1k, W2k, b2k, W1q, b1q, W2q, b2q, w);
  k_gate<<<dim3(8, B_), dim3(256), 0, stream>>>(Q, w);
  k_qbarT<<<dim3(B_ * H_), dim3(256), 0, stream>>>(WQ, WK, w);
  k_logits<<<dim3(16, B_), dim3(256), 0, stream>>>(K, w, Rbuf);
  k_softmax<<<dim3(B_ * 16), dim3(256), 0, stream>>>(Rbuf, Pbuf);
  k_pv<<<dim3(64), dim3(256), 0, stream>>>(V, Pbuf, w);
  k_oprep<<<dim3(B_), dim3(256), 0, stream>>>(WV, w);
  k_write<<<dim3(S_ / 8, B_), dim3(256), 0, stream>>>(w, out);
}